// MyConv2D_62070867361867
// MI455X (gfx1250) — compile-verified
//
#include <hip/hip_runtime.h>

typedef __attribute__((ext_vector_type(2))) float v2f;
typedef __attribute__((ext_vector_type(8))) float v8f;

#define B_N   64
#define C_IN  64
#define C_OUT 128
#define H_IN  64
#define W_IN  64
#define H_OUT 62
#define W_OUT 62
#define RED   576                 // C_IN * 3 * 3
#define NCOLS 66                  // patch cols needed: 64 valid + 2 (clamped) tail
#define PADC  67                  // padded col stride (bank spread)
#define WT_ELEMS (9 * 32 * 128)   // (tap r, ci-pair, co) float2 entries

#define HAVE_ASYNC_LDS __has_builtin(__builtin_amdgcn_global_load_async_to_lds_b32)

typedef __attribute__((address_space(1))) int gas_int;   // global (device) AS pointer pointee
typedef __attribute__((address_space(3))) int las_int;   // LDS AS pointer pointee

// One-time weight transpose: wt[(r*32 + cip)*128 + co] = (W[co][2cip][r], W[co][2cip+1][r])
__global__ void wt_transpose_kernel(const float* __restrict__ w, v2f* __restrict__ wt) {
    int id = blockIdx.x * 256 + threadIdx.x;
    if (id >= WT_ELEMS) return;
    int co  = id & 127;
    int t   = id >> 7;
    int cip = t & 31;
    int r   = t >> 5;
    int ci0 = cip * 2;
    v2f v;
    v.x = w[(size_t)co * RED + ci0 * 9 + r];
    v.y = w[(size_t)co * RED + (ci0 + 1) * 9 + r];
    wt[id] = v;
}

template <bool USE_WT>
__global__ __launch_bounds__(256) void conv_wmma_kernel(
    const float* __restrict__ x, const float* __restrict__ w,
    const v2f* __restrict__ wt, const float* __restrict__ bias,
    float* __restrict__ out)
{
    // LDS patch: [ci-pair 0..31][row 0..2][col 0..PADC-1] of float2 (ci, ci+1)
    __shared__ v2f lds2[32 * 3 * PADC];   // 51,456 B

    const int oh  = blockIdx.x;           // one full output row per block
    const int b   = blockIdx.y;
    const int tid = threadIdx.x;

    // ---- stage input patch: 64 ci x 3 rows x 66 cols -> LDS (ci-pair interleaved) ----
    const float* xb = x + (size_t)b * C_IN * H_IN * W_IN;
    float* ldsf = (float*)lds2;
    for (int idx = tid; idx < C_IN * 3 * NCOLS; idx += 256) {
        int ci   = idx / (3 * NCOLS);
        int rem  = idx - ci * (3 * NCOLS);
        int row  = rem / NCOLS;
        int col  = rem - row * NCOLS;
        int gcol = col > (W_IN - 1) ? (W_IN - 1) : col;  // cols 64,65 feed only masked pixels
        const float* gsrc = &xb[((size_t)ci * H_IN + (oh + row)) * W_IN + gcol];
        int lidx = (((ci >> 1) * 3 + row) * PADC + col) * 2 + (ci & 1);
#if HAVE_ASYNC_LDS
        __builtin_amdgcn_global_load_async_to_lds_b32(
            (gas_int*)gsrc, (las_int*)&ldsf[lidx], /*offset=*/0, /*cpol=*/0);
#else
        ldsf[lidx] = *gsrc;
#endif
    }
#if HAVE_ASYNC_LDS
#if __has_builtin(__builtin_amdgcn_s_wait_asynccnt)
    __builtin_amdgcn_s_wait_asynccnt(0);
#else
    asm volatile("s_wait_asynccnt 0x0" ::: "memory");
#endif
#endif
    __syncthreads();

    const int wave   = tid >> 5;
    const int lane   = tid & 31;
    const int px     = lane & 15;          // N (pixel) within a 16-px tile
    const int hi     = lane >> 4;          // 0: K=0,1 / M=vgpr ; 1: K=2,3 / M=vgpr+8
    const int cobase = wave * 16;

    v8f acc0 = {}, acc1 = {}, acc2 = {}, acc3 = {};
    const float* wrow = w + (size_t)(cobase + px) * RED;   // fallback A row (co = cobase+px)

    for (int r = 0; r < 9; ++r) {
        const int kh = r / 3;
        const int kw = r - kh * 3;
        const v2f* wtp  = USE_WT ? (wt + ((size_t)(r * 32) << 7) + cobase + px) : nullptr;
        const v2f* brow = &lds2[kh * PADC + px + kw];      // + cip*3*PADC + 16*tile
        #pragma unroll
        for (int cb = 0; cb < 16; ++cb) {  // 16 steps of K=4 over 64 input channels
            const int cip = cb * 2 + hi;   // ci pair; ci0 = 2*cip
            v2f a;
            if (USE_WT) {
                a = wtp[(size_t)cip << 7]; // coalesced global_load_b64
            } else {
                const int ci0 = cip * 2;
                a.x = wrow[ci0 * 9 + r];
                a.y = wrow[(ci0 + 1) * 9 + r];
            }
            const v2f* bp = brow + cip * (3 * PADC);       // ds_load_b64 x4, one A reuse
            acc0 = __builtin_amdgcn_wmma_f32_16x16x4_f32(false, a, false, bp[0],  (short)0, acc0, false, false);
            acc1 = __builtin_amdgcn_wmma_f32_16x16x4_f32(false, a, false, bp[16], (short)0, acc1, false, false);
            acc2 = __builtin_amdgcn_wmma_f32_16x16x4_f32(false, a, false, bp[32], (short)0, acc2, false, false);
            acc3 = __builtin_amdgcn_wmma_f32_16x16x4_f32(false, a, false, bp[48], (short)0, acc3, false, false);
        }
    }

    // ---- epilogue: bias + predicated stores (after all WMMAs; EXEC full during matrix ops)
    float* obase = out + (size_t)b * C_OUT * H_OUT * W_OUT + (size_t)oh * W_OUT;
    v8f accs[4] = {acc0, acc1, acc2, acc3};
    #pragma unroll
    for (int t = 0; t < 4; ++t) {
        const int ow = t * 16 + px;
        if (ow < W_OUT) {
            #pragma unroll
            for (int i = 0; i < 8; ++i) {
                const int co = cobase + i + 8 * hi;   // C/D layout: vgpr i, M = i + 8*hi
                obase[(size_t)co * H_OUT * W_OUT + ow] = accs[t][i] + bias[co];
            }
        }
    }
}

extern "C" void kernel_launch(void* const* d_in, const int* in_sizes, int n_in,
                              void* d_out, int out_size, void* d_ws, size_t ws_size,
                              hipStream_t stream) {
    const float* x    = (const float*)d_in[0];
    const float* w    = (const float*)d_in[1];
    const float* bias = (const float*)d_in[2];
    float* out        = (float*)d_out;

    dim3 grid(H_OUT, B_N);                                   // 62 x 64 blocks
    const size_t wt_bytes = (size_t)WT_ELEMS * sizeof(v2f);  // 294,912 B

    if (ws_size >= wt_bytes) {
        v2f* wt = (v2f*)d_ws;
        wt_transpose_kernel<<<(WT_ELEMS + 255) / 256, 256, 0, stream>>>(w, wt);
        conv_wmma_kernel<true><<<grid, 256, 0, stream>>>(x, w, wt, bias, out);
    } else {
        conv_wmma_kernel<false><<<grid, 256, 0, stream>>>(x, w, nullptr, bias, out);
    }
}